// EncodingInputLayer_78288663871628
// MI455X (gfx1250) — compile-verified
//
#include <hip/hip_runtime.h>

// ---------------- problem constants ----------------
#define BATCH      2048
#define N_OH       20
#define N_OT       23
#define NCLS       10
#define SUMC       (N_OH * NCLS)     // 200
#define EMB        9
#define OUTC       32
#define NIN        (N_OH + N_OT)     // 43 input features per batch row
#define PIX        2304              // 48*48
#define V_MOD      (BATCH * EMB)     // 18432  (v.flat tiling period)
#define O_MOD      (BATCH * N_OT)    // 47104  (others.flat tiling period)
#define V_BSTRIDE  (EMB  * PIX)      // 20736  (per-batch advance in v flat space)
#define O_BSTRIDE  (N_OT * PIX)      // 52992  (per-batch advance in o flat space)
#define KP         36                // padded K: 9 v + 3 zero + 23 o + 1 zero
#define NCHUNK     (KP / 4)          // 9 WMMA K-chunks
#define VEXT_N     46080             // covers base_v(<18432)+p(<2304)+11*2304
#define OEXT_N     102400            // covers base_o(<47104)+p(<2304)+23*2304

// ---------------- workspace layout (float units) ----------------
#define WS_V     0
#define WS_VEXT  (WS_V    + V_MOD)          // 18432
#define WS_OEXT  (WS_VEXT + VEXT_N)         // 64512
#define WS_WC    (WS_OEXT + OEXT_N)         // 166912
#define WS_BIAS  (WS_WC   + OUTC * KP)      // 168064
// total: 168096 floats = ~672 KB

typedef __attribute__((ext_vector_type(2))) float v2f;
typedef __attribute__((ext_vector_type(8))) float v8f;

// ---------------------------------------------------------------------------
// Kernel 1: v[b,e] (embedding gather-sum), composite weights Wc(32x36), bias.
// ---------------------------------------------------------------------------
__global__ __launch_bounds__(256) void setup1_kernel(
    const float* __restrict__ x,
    const float* __restrict__ fc_w, const float* __restrict__ fc_b,
    const float* __restrict__ oh_w, const float* __restrict__ oh_b,
    const float* __restrict__ ot_w, const float* __restrict__ ot_b,
    const float* __restrict__ all_w, const float* __restrict__ all_b,
    float* __restrict__ ws)
{
    const int gid = blockIdx.x * blockDim.x + threadIdx.x;
    if (gid < BATCH) {
        // v[gid, 0..8] = fc_b + sum_j fc_w[:, j*10 + idx]
        const float* xr = x + gid * NIN;
        float acc[EMB];
        #pragma unroll
        for (int e = 0; e < EMB; ++e) acc[e] = fc_b[e];
        for (int j = 0; j < N_OH; ++j) {
            const int c = (int)xr[j];
            const int col = j * NCLS + c;
            #pragma unroll
            for (int e = 0; e < EMB; ++e) acc[e] += fc_w[e * SUMC + col];
        }
        #pragma unroll
        for (int e = 0; e < EMB; ++e) ws[WS_V + gid * EMB + e] = acc[e];
    } else if (gid < BATCH + OUTC * KP) {
        // composite weight Wc[o,k]
        const int t = gid - BATCH;
        const int o = t / KP;
        const int k = t % KP;
        float w = 0.0f;
        if (k < EMB) {                       // v path: all_w[:, :9] @ oh_w
            for (int m = 0; m < EMB; ++m) w += all_w[o * OUTC + m] * oh_w[m * EMB + k];
        } else if (k >= 12 && k < 12 + N_OT) { // o path: all_w[:, 9:] @ ot_w
            const int j = k - 12;
            for (int m = 0; m < N_OT; ++m) w += all_w[o * OUTC + EMB + m] * ot_w[m * N_OT + j];
        }                                     // k in {9,10,11,35}: zero pad
        ws[WS_WC + o * KP + k] = w;
    } else if (gid < BATCH + OUTC * KP + OUTC) {
        // fused bias
        const int o = gid - (BATCH + OUTC * KP);
        float bsum = all_b[o];
        for (int m = 0; m < EMB; ++m)  bsum += all_w[o * OUTC + m] * oh_b[m];
        for (int m = 0; m < N_OT; ++m) bsum += all_w[o * OUTC + EMB + m] * ot_b[m];
        ws[WS_BIAS + o] = bsum;
    }
}

// ---------------------------------------------------------------------------
// Kernel 2: wrap-extended gather tables (kills modular arithmetic in hot loop)
//   vext[t] = v.flat[t % 18432]       t < 46080
//   oext[t] = others.flat[t % 47104]  t < 102400, others.flat[s] = x[(s/23)*43 + 20 + s%23]
// ---------------------------------------------------------------------------
__global__ __launch_bounds__(256) void setup2_kernel(
    const float* __restrict__ x, float* __restrict__ ws)
{
    const int gid = blockIdx.x * blockDim.x + threadIdx.x;
    if (gid < VEXT_N) {
        ws[WS_VEXT + gid] = ws[WS_V + (gid % V_MOD)];
    } else if (gid < VEXT_N + OEXT_N) {
        const int t = gid - VEXT_N;
        const int s = t % O_MOD;
        ws[WS_OEXT + t] = x[(s / N_OT) * NIN + N_OH + (s % N_OT)];
    }
}

// ---------------------------------------------------------------------------
// Kernel 3: main WMMA kernel.
// Block = 256 threads (8 waves), one batch b per block.
// Each wave: D-tile = 16 pixels x 32 out-channels via 2 accumulators,
// K=36 as 9 chunks of V_WMMA_F32_16X16X4_F32 -> 18 WMMAs per tile.
// ---------------------------------------------------------------------------
__global__ __launch_bounds__(256) void fused_wmma_kernel(
    const float* __restrict__ ws, float* __restrict__ out)
{
    const int lane = threadIdx.x & 31;
    const int wid  = threadIdx.x >> 5;
    const int n    = lane & 15;       // column / pixel-in-tile
    const int half = lane >> 4;       // 0: lanes 0-15, 1: lanes 16-31
    const int b    = blockIdx.x;

    const float* __restrict__ vext = ws + WS_VEXT;
    const float* __restrict__ oext = ws + WS_OEXT;
    const float* __restrict__ Wc   = ws + WS_WC;
    const float* __restrict__ bias = ws + WS_BIAS;

    const int base_v = (b * V_BSTRIDE) % V_MOD;   // scalar (uniform)
    const int base_o = (b * O_BSTRIDE) % O_MOD;

    // --- loop-invariant A operands (composite weights), ISA 16x4 f32 layout:
    // lane L: VGPR0 = A[M=L%16][K = 4c + 2*half], VGPR1 = K+1
    v2f A[2][NCHUNK];
    {
        const int row0 = n, row1 = 16 + n;
        #pragma unroll
        for (int c = 0; c < NCHUNK; ++c) {
            const int kA = 4 * c + 2 * half;
            A[0][c].x = Wc[row0 * KP + kA];
            A[0][c].y = Wc[row0 * KP + kA + 1];
            A[1][c].x = Wc[row1 * KP + kA];
            A[1][c].y = Wc[row1 * KP + kA + 1];
        }
    }

    // --- loop-invariant C init (fused bias); D layout: VGPR r -> M = r + 8*half
    v8f Cinit[2];
    #pragma unroll
    for (int r = 0; r < 8; ++r) {
        Cinit[0][r] = bias[r + half * 8];
        Cinit[1][r] = bias[16 + r + half * 8];
    }

    // per-lane gather bases (fold the half-dependent K offset in once)
    const int vb = base_v + n + half * 2 * PIX;
    const int ob = base_o + n + half * 2 * PIX;
    // per-lane store base: channel (8*half), pixel n
    float* __restrict__ outLane = out + (size_t)b * (OUTC * PIX) + half * 8 * PIX + n;

    for (int t = wid; t < PIX / 16; t += 8) {
        const int p0  = t * 16;
        const int svh = vb + p0;
        const int soh = ob + p0;

        // --- gather B operands (16 pixels x K=36), all-constant immediate offsets
        v2f Bm[NCHUNK];
        #pragma unroll
        for (int c = 0; c < NCHUNK; ++c) {
            if (c < 3) {              // K rows 0..11 : v table (9 real + 3 zero-wt pad)
                Bm[c].x = vext[svh + (4 * c) * PIX];
                Bm[c].y = vext[svh + (4 * c + 1) * PIX];
            } else {                  // K rows 12..35 : o table (23 real + 1 zero-wt pad)
                Bm[c].x = oext[soh + (4 * c - 12) * PIX];
                Bm[c].y = oext[soh + (4 * c - 11) * PIX];
            }
        }

        // --- 18x V_WMMA_F32_16X16X4_F32
        v8f acc0 = Cinit[0];
        v8f acc1 = Cinit[1];
        #pragma unroll
        for (int c = 0; c < NCHUNK; ++c) {
            acc0 = __builtin_amdgcn_wmma_f32_16x16x4_f32(
                false, A[0][c], false, Bm[c], (short)0, acc0, false, false);
            acc1 = __builtin_amdgcn_wmma_f32_16x16x4_f32(
                false, A[1][c], false, Bm[c], (short)0, acc1, false, false);
        }

        // --- scatter D: lane holds channel (r + 8*half [+16]) at pixel p0+n
        #pragma unroll
        for (int r = 0; r < 8; ++r) {
            outLane[r * PIX + p0]        = acc0[r];
            outLane[(16 + r) * PIX + p0] = acc1[r];
        }
    }
}

// ---------------------------------------------------------------------------
extern "C" void kernel_launch(void* const* d_in, const int* in_sizes, int n_in,
                              void* d_out, int out_size, void* d_ws, size_t ws_size,
                              hipStream_t stream) {
    const float* x     = (const float*)d_in[0];
    const float* fc_w  = (const float*)d_in[1];
    const float* fc_b  = (const float*)d_in[2];
    const float* oh_w  = (const float*)d_in[3];
    const float* oh_b  = (const float*)d_in[4];
    const float* ot_w  = (const float*)d_in[5];
    const float* ot_b  = (const float*)d_in[6];
    const float* all_w = (const float*)d_in[7];
    const float* all_b = (const float*)d_in[8];
    float* ws  = (float*)d_ws;
    float* out = (float*)d_out;

    const int work1 = BATCH + OUTC * KP + OUTC;            // 3232
    setup1_kernel<<<(work1 + 255) / 256, 256, 0, stream>>>(
        x, fc_w, fc_b, oh_w, oh_b, ot_w, ot_b, all_w, all_b, ws);

    const int work2 = VEXT_N + OEXT_N;                     // 148480
    setup2_kernel<<<(work2 + 255) / 256, 256, 0, stream>>>(x, ws);

    fused_wmma_kernel<<<BATCH, 256, 0, stream>>>(ws, out);
}